// LightGCN_6365141532768
// MI455X (gfx1250) — compile-verified
//
#include <hip/hip_runtime.h>
#include <stdint.h>

#define NUM_USERS 500000
#define NUM_ITEMS 100000
#define DIM 32
#define NUM_LAYERS 3
#define NTOT (NUM_USERS + NUM_ITEMS)
#define E_BIP 1000000

#define CHUNK 512                              /* edges per LDS chunk       */
#define NCHUNK ((E_BIP + CHUNK - 1) / CHUNK)   /* 1954 chunks               */
#define SCAT_THREADS 512                       /* 16 waves per block        */
#define CHUNKS_PER_BLOCK 4

typedef unsigned int u32x4 __attribute__((ext_vector_type(4)));
typedef int          i32x4 __attribute__((ext_vector_type(4)));
typedef int          i32x8 __attribute__((ext_vector_type(8)));

#if defined(__has_builtin)
#  if __has_builtin(__builtin_amdgcn_tensor_load_to_lds)
#    define USE_TDM 1
#  endif
#endif

__device__ __forceinline__ float wave_rowsum(float v) {
  v += __shfl_xor(v, 16, 32);
  v += __shfl_xor(v, 8, 32);
  v += __shfl_xor(v, 4, 32);
  v += __shfl_xor(v, 2, 32);
  v += __shfl_xor(v, 1, 32);
  return v;
}

#ifdef USE_TDM
// 1D TDM tile: n_dwords contiguous 4-byte elements, global -> LDS.
// D# packing per CDNA5 ISA 8.3/8.4: group0 {count=1, lds_addr, global_addr[56:0], type=2},
// group1 {data_size=4B, tensor_dim0=n, tensor_dim1=1, tile_dim0=n, stride0=n},
// groups 2/3 benign (dims=1, tiles unused).
__device__ __forceinline__ void tdm_load_1d(void* lds_ptr, const void* gptr, int n_dwords) {
  unsigned lds = (unsigned)(unsigned long long)lds_ptr; // flat addr low 32 bits == LDS offset
  unsigned long long ga = (unsigned long long)gptr;
  unsigned td0 = (unsigned)n_dwords;

  u32x4 g0 = {};
  g0[0] = 1u;                                                   // count=1, user mode
  g0[1] = lds;                                                  // lds_addr
  g0[2] = (unsigned)(ga & 0xFFFFFFFFull);                       // global_addr[31:0]
  g0[3] = (unsigned)((ga >> 32) & 0x01FFFFFFull) | 0x80000000u; // addr[56:32] | type=2

  i32x8 g1 = {};
  g1[0] = 0x00020000;                                  // data_size=2 (4 bytes)
  g1[1] = (int)((td0 & 0xFFFFu) << 16);                // tensor_dim0[15:0] @ bits63:48
  g1[2] = (int)(((td0 >> 16) & 0xFFFFu) | (1u << 16)); // tensor_dim0 hi | tensor_dim1=1
  g1[3] = (int)((td0 & 0xFFFFu) << 16);                // tile_dim0 = n
  g1[4] = 0;                                           // tile_dim1/2 unused
  g1[5] = (int)td0;                                    // tensor_dim0_stride
  g1[6] = 0;
  g1[7] = 0;

  i32x4 g2 = {};
  g2[0] = 1;          // tensor_dim2 = 1 (safe)
  g2[1] = 1;          // tensor_dim3 = 1
  i32x4 g3 = {};
  g3[1] = 0x00010000; // tensor_dim4 = 1

#if __clang_major__ >= 23
  i32x8 g4 = {};
  __builtin_amdgcn_tensor_load_to_lds(g0, g1, g2, g3, g4, 0);
#else
  __builtin_amdgcn_tensor_load_to_lds(g0, g1, g2, g3, 0);
#endif
}

__device__ __forceinline__ void tdm_issue_chunk(int c, int* s_src, int* s_dst, float* s_w,
                                                const int* esrc, const int* edst,
                                                const float* ew) {
  int base = c * CHUNK;
  int cnt = E_BIP - base;
  if (cnt > CHUNK) cnt = CHUNK;
  tdm_load_1d(s_src, esrc + base, cnt);
  tdm_load_1d(s_dst, edst + base, cnt);
  tdm_load_1d(s_w, ew + base, cnt);
}
#endif

// --- Kernel 1: x0 = [l2norm(user); l2norm(audio + 0.5*(artist+album))],
//     seed acc = x0, zero first-layer destination. One wave per row (lane = dim).
__global__ void __launch_bounds__(256) init_x(
    const float* __restrict__ user_w, const float* __restrict__ audio,
    const float* __restrict__ artist_w, const float* __restrict__ album_w,
    const int* __restrict__ artist_ids, const int* __restrict__ album_ids,
    float* __restrict__ x, float* __restrict__ acc, float* __restrict__ y) {
  int gtid = blockIdx.x * blockDim.x + threadIdx.x;
  int row = gtid >> 5;
  int lane = gtid & 31;
  if (row >= NTOT) return;
  float v;
  if (row < NUM_USERS) {
    v = user_w[(size_t)row * DIM + lane];
  } else {
    int it = row - NUM_USERS;
    int ar = artist_ids[it];
    int al = album_ids[it];
    v = audio[(size_t)it * DIM + lane] +
        0.5f * (artist_w[(size_t)ar * DIM + lane] + album_w[(size_t)al * DIM + lane]);
  }
  float n = sqrtf(wave_rowsum(v * v));
  v = v / fmaxf(n, 1e-12f);
  size_t o = (size_t)row * DIM + lane;
  x[o] = v;
  acc[o] = v;
  y[o] = 0.0f;
}

// --- Kernel 2: one propagation layer. Double-buffered TDM pipeline:
//     wave 0 issues async edge-stream DMA for chunk i+1 while all waves run
//     wave-per-edge gather + per-lane f32 atomic scatter on chunk i.
//     TENSORcnt is in-order per wave, so s_wait_tensorcnt<=3 retires exactly
//     the current chunk's 3 tensor loads while next chunk's 3 stay in flight.
__global__ void __launch_bounds__(SCAT_THREADS) scatter_layer(
    const int* __restrict__ esrc, const int* __restrict__ edst,
    const float* __restrict__ ew,
    const float* __restrict__ x, float* __restrict__ y) {
  __shared__ int s_src[2][CHUNK];
  __shared__ int s_dst[2][CHUNK];
  __shared__ float s_w[2][CHUNK];

  const int wave = threadIdx.x >> 5;
  const int lane = threadIdx.x & 31;
  const int nw = SCAT_THREADS >> 5;
  const int c0 = blockIdx.x;
  const int stride = gridDim.x;
  const bool issuer = (wave == 0);

#ifdef USE_TDM
  if (c0 < NCHUNK && issuer)
    tdm_issue_chunk(c0, s_src[0], s_dst[0], s_w[0], esrc, edst, ew);
#endif

  int buf = 0;
  for (int c = c0; c < NCHUNK; c += stride, buf ^= 1) {
    const int nc = c + stride;
    const bool has_next = (nc < NCHUNK);
    const int base = c * CHUNK;
    int cnt = E_BIP - base;
    if (cnt > CHUNK) cnt = CHUNK;

#ifdef USE_TDM
    if (issuer) {
      if (has_next)
        tdm_issue_chunk(nc, s_src[buf ^ 1], s_dst[buf ^ 1], s_w[buf ^ 1], esrc, edst, ew);
      if (has_next)
        __builtin_amdgcn_s_wait_tensorcnt(3);  // current chunk done, next in flight
      else
        __builtin_amdgcn_s_wait_tensorcnt(0);
    }
#else
    for (int e = threadIdx.x; e < cnt; e += blockDim.x) {
      s_src[buf][e] = esrc[base + e];
      s_dst[buf][e] = edst[base + e];
      s_w[buf][e] = ew[base + e];
    }
#endif
    __syncthreads();

    for (int e = wave; e < cnt; e += nw) {
      int u = s_src[buf][e];               // LDS broadcast across the wave
      int it = s_dst[buf][e] + NUM_USERS;
      float w = s_w[buf][e];
      float xu = x[(size_t)u * DIM + lane];   // coalesced 128B row gathers (L2-resident)
      float xi = x[(size_t)it * DIM + lane];
      atomicAdd(&y[(size_t)it * DIM + lane], w * xu);
      atomicAdd(&y[(size_t)u * DIM + lane], w * xi);
    }
    __syncthreads();  // buffer free before it is overwritten next iteration
  }
}

// --- Kernel 3: acc += y, and zero the buffer that becomes next layer's y.
__global__ void __launch_bounds__(256) combine(
    float4* __restrict__ acc, const float4* __restrict__ y,
    float4* __restrict__ zbuf, int n4) {
  int i = blockIdx.x * blockDim.x + threadIdx.x;
  if (i >= n4) return;
  float4 a = acc[i];
  float4 b = y[i];
  a.x += b.x; a.y += b.y; a.z += b.z; a.w += b.w;
  acc[i] = a;
  if (zbuf) {
    float4 z = {0.0f, 0.0f, 0.0f, 0.0f};
    zbuf[i] = z;
  }
}

// --- Kernel 4: out = l2norm(acc/4) in place; append align_loss = 0.
__global__ void __launch_bounds__(256) finalize(float* __restrict__ out) {
  int gtid = blockIdx.x * blockDim.x + threadIdx.x;
  int row = gtid >> 5;
  int lane = gtid & 31;
  if (row >= NTOT) return;
  size_t o = (size_t)row * DIM + lane;
  float v = out[o] * 0.25f;
  float n = sqrtf(wave_rowsum(v * v));
  out[o] = v / fmaxf(n, 1e-12f);
  if (gtid == 0) out[(size_t)NTOT * DIM] = 0.0f;  // align_loss
}

extern "C" void kernel_launch(void* const* d_in, const int* in_sizes, int n_in,
                              void* d_out, int out_size, void* d_ws, size_t ws_size,
                              hipStream_t stream) {
  (void)in_sizes; (void)n_in; (void)out_size; (void)ws_size;
  const float* user_w     = (const float*)d_in[0];
  const float* audio      = (const float*)d_in[1];
  const float* artist_w   = (const float*)d_in[2];
  const float* album_w    = (const float*)d_in[3];
  const float* ew         = (const float*)d_in[4];
  const int*   artist_ids = (const int*)d_in[5];
  const int*   album_ids  = (const int*)d_in[6];
  const int*   esrc       = (const int*)d_in[7];
  const int*   edst       = (const int*)d_in[8];

  float* acc  = (float*)d_out;                 // 600k x 32 accumulator lives in d_out
  float* bufA = (float*)d_ws;                  // 76.8 MB
  float* bufB = bufA + (size_t)NTOT * DIM;     // 76.8 MB (x+y both L2-resident)

  const int rowThreads = NTOT * DIM;           // 19.2M lanes, one per element
  const int rowBlocks  = (rowThreads + 255) / 256;
  const int scatBlocks = (NCHUNK + CHUNKS_PER_BLOCK - 1) / CHUNKS_PER_BLOCK; // 489
  const int n4         = NTOT * DIM / 4;
  const int cBlocks    = (n4 + 255) / 256;

  init_x<<<rowBlocks, 256, 0, stream>>>(user_w, audio, artist_w, album_w,
                                        artist_ids, album_ids, bufA, acc, bufB);

  float* x = bufA;
  float* y = bufB;
  for (int l = 0; l < NUM_LAYERS; ++l) {
    scatter_layer<<<scatBlocks, SCAT_THREADS, 0, stream>>>(esrc, edst, ew, x, y);
    float* z = (l == NUM_LAYERS - 1) ? nullptr : x;  // x becomes next layer's y
    combine<<<cBlocks, 256, 0, stream>>>((float4*)acc, (const float4*)y, (float4*)z, n4);
    float* t = x; x = y; y = t;
  }

  finalize<<<rowBlocks, 256, 0, stream>>>(acc);
}